// GCN6_20693152432417
// MI455X (gfx1250) — compile-verified
//
#include <hip/hip_runtime.h>

#define N_NODES  50000
#define N_EDGES  800000
#define N_GRAPHS 1000
#define F_IN     128
#define HID      64

typedef __attribute__((ext_vector_type(2))) float v2f;
typedef __attribute__((ext_vector_type(8))) float v8f;

#define AS3 __attribute__((address_space(3)))

// ---- async global -> LDS copy (ASYNCcnt-tracked), per CDNA5 ISA ----
// global_load_async_to_lds_b128: VDST = LDS byte address, VADDR = 64-bit global address.
__device__ __forceinline__ void async_g2l_b128(const float* gsrc, AS3 float* ldst) {
  unsigned long long ga = (unsigned long long)(__UINTPTR_TYPE__)gsrc;
  unsigned la = (unsigned)(__UINTPTR_TYPE__)ldst;
  asm volatile("global_load_async_to_lds_b128 %0, %1, off" :: "v"(la), "v"(ga) : "memory");
}

__device__ __forceinline__ void wait_async0() {
  asm volatile("s_wait_asynccnt 0x0" ::: "memory");
}

// ---------------- small utility kernels ----------------

__global__ void k_fill(float* __restrict__ p, float v, int n) {
  int i = blockIdx.x * blockDim.x + threadIdx.x;
  if (i < n) p[i] = v;
}

__global__ void k_deg_accum(float* __restrict__ deg, const int* __restrict__ dst, int e) {
  int i = blockIdx.x * blockDim.x + threadIdx.x;
  if (i < e) atomicAdd(&deg[dst[i]], 1.0f);
}

__global__ void k_rsqrt_inplace(float* __restrict__ d, int n) {
  int i = blockIdx.x * blockDim.x + threadIdx.x;
  if (i < n) d[i] = rsqrtf(d[i]);
}

__global__ void k_enorm(float* __restrict__ enorm, const float* __restrict__ dis,
                        const int* __restrict__ src, const int* __restrict__ dst, int e) {
  int i = blockIdx.x * blockDim.x + threadIdx.x;
  if (i < e) enorm[i] = dis[src[i]] * dis[dst[i]];
}

// ---------------- WMMA f32 GEMM: H[M x 64] = X[M x K] @ W[K x 64] ----------------
// Block = 128 threads (4 waves); block handles one 16-row strip of X, wave w the
// 16-col tile w. The 16xK A-strip is contiguous in memory -> staged into LDS once
// via async B128 copies (coalesced), then WMMA A-fragments come from LDS.
// V_WMMA_F32_16X16X4_F32 lane layout (ISA 7.12.2):
//   A 16x4: lanes 0-15 K={k,k+1}, lanes 16-31 K={k+2,k+3}, row M = lane&15
//   B 4x16: row striped across 16 lanes per half-VGPR, same K mirroring
//   C/D 16x16: VGPR j = row (j + 8*(lane>>4)), col = lane&15
template<int K>
__global__ void k_gemm_wmma(const float* __restrict__ X, const float* __restrict__ W,
                            float* __restrict__ H) {
  __shared__ float tileA[16 * K];

  const int tid  = threadIdx.x;
  const int wave = tid >> 5;
  const int lane = tid & 31;
  const int m    = lane & 15;
  const int hi   = lane >> 4;
  const int row0 = blockIdx.x * 16;   // M is an exact multiple of 16
  const int col0 = wave * 16;

  // cooperative async stage: 16*K contiguous floats starting at X + row0*K
  {
    AS3 float* lbase = (AS3 float*)tileA;
    const float* gbase = X + (long)row0 * K;
    const int chunks = (16 * K) / 4;  // float4 chunks
#pragma unroll
    for (int c = tid; c < chunks; c += 128) {
      async_g2l_b128(gbase + 4 * c, lbase + 4 * c);
    }
    wait_async0();
    __syncthreads();
  }

  v8f acc = {};
  const float* arow = tileA + m * K;
  for (int k = 0; k < K; k += 4) {
    v2f a, b;
    a.x = arow[k + 2 * hi];
    a.y = arow[k + 2 * hi + 1];
    b.x = W[(long)(k + 2 * hi) * HID + col0 + m];
    b.y = W[(long)(k + 2 * hi + 1) * HID + col0 + m];
    acc = __builtin_amdgcn_wmma_f32_16x16x4_f32(
        /*neg_a=*/false, a, /*neg_b=*/false, b,
        /*c_mod=*/(short)0, acc, /*reuse_a=*/false, /*reuse_b=*/false);
  }
#pragma unroll
  for (int j = 0; j < 8; ++j) {
    H[(long)(row0 + j + 8 * hi) * HID + col0 + m] = acc[j];
  }
}

// ---------------- edge aggregation: agg[dst] += h[src] * enorm ----------------
// One thread handles 2 consecutive features of one edge (wave32 covers one full
// 64-feature edge): coalesced b64 gather of h[src], coalesced f32 atomic scatter.
__global__ void k_edge_agg(float* __restrict__ agg, const float* __restrict__ h,
                           const float* __restrict__ enorm,
                           const int* __restrict__ src, const int* __restrict__ dst) {
  const long total = (long)N_EDGES * 32;
  for (long t = (long)blockIdx.x * blockDim.x + threadIdx.x; t < total;
       t += (long)gridDim.x * blockDim.x) {
    const int e  = (int)(t >> 5);
    const int fp = ((int)t & 31) * 2;
    const float en = enorm[e];
    const float2 hv = *(const float2*)(h + (long)src[e] * HID + fp);
    float* ap = agg + (long)dst[e] * HID + fp;
    atomicAdd(ap,     hv.x * en);
    atomicAdd(ap + 1, hv.y * en);
  }
}

// ---------------- epilogue: hout = relu(agg + hin * dis^2 + b) ----------------
__global__ void k_finish(float* __restrict__ hout, const float* __restrict__ agg,
                         const float* __restrict__ hin, const float* __restrict__ dis,
                         const float* __restrict__ b) {
  const long total = (long)N_NODES * HID;
  long t = (long)blockIdx.x * blockDim.x + threadIdx.x;
  if (t >= total) return;
  const int v = (int)(t >> 6);
  const int f = (int)(t & 63);
  const float s = dis[v];
  const float r = agg[t] + hin[t] * s * s + b[f];
  hout[t] = r > 0.0f ? r : 0.0f;
}

// ---------------- pooling ----------------
__global__ void k_counts(float* __restrict__ counts, const int* __restrict__ batch) {
  int i = blockIdx.x * blockDim.x + threadIdx.x;
  if (i < N_NODES) atomicAdd(&counts[batch[i]], 1.0f);
}

__global__ void k_pool(float* __restrict__ pooled, const float* __restrict__ h,
                       const int* __restrict__ batch) {
  const long total = (long)N_NODES * HID;
  long t = (long)blockIdx.x * blockDim.x + threadIdx.x;
  if (t >= total) return;
  const int v = (int)(t >> 6);
  const int f = (int)(t & 63);
  atomicAdd(&pooled[(long)batch[v] * HID + f], h[t]);
}

// one wave per graph: out[g] = dot(pooled[g]/max(cnt,1), Wl) + bl
__global__ void k_final(float* __restrict__ out, const float* __restrict__ pooled,
                        const float* __restrict__ counts, const float* __restrict__ Wl,
                        const float* __restrict__ bl) {
  const int g = (blockIdx.x * blockDim.x + threadIdx.x) >> 5;
  const int lane = threadIdx.x & 31;
  if (g >= N_GRAPHS) return;
  const float inv = 1.0f / fmaxf(counts[g], 1.0f);
  float s = pooled[(long)g * HID + lane] * Wl[lane] +
            pooled[(long)g * HID + 32 + lane] * Wl[32 + lane];
#pragma unroll
  for (int off = 16; off > 0; off >>= 1) s += __shfl_xor(s, off, 32);
  if (lane == 0) out[g] = s * inv + bl[0];
}

// ---------------- host-side launch ----------------
extern "C" void kernel_launch(void* const* d_in, const int* in_sizes, int n_in,
                              void* d_out, int out_size, void* d_ws, size_t ws_size,
                              hipStream_t stream) {
  const float* x   = (const float*)d_in[0];
  const float* W1  = (const float*)d_in[1];
  const float* b1  = (const float*)d_in[2];
  const float* W2  = (const float*)d_in[3];
  const float* b2  = (const float*)d_in[4];
  const float* Wl  = (const float*)d_in[5];
  const float* bl  = (const float*)d_in[6];
  const int*   ei  = (const int*)d_in[7];     // [2, E]
  const int*   bat = (const int*)d_in[8];     // [N]
  float* out = (float*)d_out;

  const int* src = ei;
  const int* dst = ei + N_EDGES;

  // workspace layout (floats)
  float* ws     = (float*)d_ws;
  float* dis    = ws;                                  // N
  float* enorm  = dis + N_NODES;                       // E
  float* hA     = enorm + N_EDGES;                     // N*HID
  float* hB     = hA + (long)N_NODES * HID;            // N*HID
  float* agg    = hB + (long)N_NODES * HID;            // N*HID
  float* pooled = agg + (long)N_NODES * HID;           // G*HID
  float* counts = pooled + (long)N_GRAPHS * HID;       // G

  const int T = 256;
  const int gN   = (N_NODES + T - 1) / T;
  const int gE   = (N_EDGES + T - 1) / T;
  const int gNH  = (int)(((long)N_NODES * HID + T - 1) / T);
  const int gGH  = (N_GRAPHS * HID + T - 1) / T;
  const int gAgg = 8192;  // grid-stride over E*32

  // degree / normalization
  k_fill<<<gN, T, 0, stream>>>(dis, 1.0f, N_NODES);
  k_deg_accum<<<gE, T, 0, stream>>>(dis, dst, N_EDGES);
  k_rsqrt_inplace<<<gN, T, 0, stream>>>(dis, N_NODES);
  k_enorm<<<gE, T, 0, stream>>>(enorm, dis, src, dst, N_EDGES);

  const int mtiles = N_NODES / 16;  // 3125 exactly

  // layer 1: hA = x @ W1 ; hB = relu(agg + hA*dis^2 + b1)
  k_gemm_wmma<F_IN><<<mtiles, 128, 0, stream>>>(x, W1, hA);
  k_fill<<<gNH, T, 0, stream>>>(agg, 0.0f, (int)((long)N_NODES * HID));
  k_edge_agg<<<gAgg, T, 0, stream>>>(agg, hA, enorm, src, dst);
  k_finish<<<gNH, T, 0, stream>>>(hB, agg, hA, dis, b1);

  // layers 2..5: shared W2/b2
  for (int l = 0; l < 4; ++l) {
    k_gemm_wmma<HID><<<mtiles, 128, 0, stream>>>(hB, W2, hA);
    k_fill<<<gNH, T, 0, stream>>>(agg, 0.0f, (int)((long)N_NODES * HID));
    k_edge_agg<<<gAgg, T, 0, stream>>>(agg, hA, enorm, src, dst);
    k_finish<<<gNH, T, 0, stream>>>(hB, agg, hA, dis, b2);
  }

  // global mean pool + readout
  k_fill<<<gGH, T, 0, stream>>>(pooled, 0.0f, N_GRAPHS * HID);
  k_fill<<<(N_GRAPHS + T - 1) / T, T, 0, stream>>>(counts, 0.0f, N_GRAPHS);
  k_counts<<<gN, T, 0, stream>>>(counts, bat);
  k_pool<<<gNH, T, 0, stream>>>(pooled, hB, bat);
  k_final<<<(N_GRAPHS * 32 + 127) / 128, 128, 0, stream>>>(out, pooled, counts, Wl, bl);
}